// GPSGraph_84533546319965
// MI455X (gfx1250) — compile-verified
//
#include <hip/hip_runtime.h>
#include <hip/hip_bf16.h>
#include <math.h>

// ---------------------------------------------------------------------------
// GCN on MI455X (gfx1250, wave32).
//   conv: H = X @ W (WMMA f32 16x16x4), then edge scatter with
//   norm = dis[src]*dis[dst] via global_atomic_add_f32, then fused
//   self-loop + bias + ReLU. Pool via atomics, tiny head + log_softmax.
// ---------------------------------------------------------------------------

typedef float v2f __attribute__((ext_vector_type(2)));
typedef float v8f __attribute__((ext_vector_type(8)));

#define N_NODES   50000
#define N_EDGES   640000
#define F_DIM     128
#define N_CLASSES 16
#define N_GRAPHS  500

// ---------------- degree / normalization -----------------------------------

__global__ void init_deg_kernel(float* __restrict__ deg, int n) {
    int i = blockIdx.x * blockDim.x + threadIdx.x;
    if (i < n) deg[i] = 1.0f;   // self-loop contribution
}

__global__ void deg_edges_kernel(const int* __restrict__ dst, float* __restrict__ deg, int e) {
    int i = blockIdx.x * blockDim.x + threadIdx.x;
    if (i < e) atomicAdd(&deg[dst[i]], 1.0f);
}

__global__ void dis_kernel(const float* __restrict__ deg, float* __restrict__ dis, int n) {
    int i = blockIdx.x * blockDim.x + threadIdx.x;
    if (i < n) dis[i] = rsqrtf(deg[i]);   // deg >= 1 always (self-loops)
}

// ---------------- generic zero ----------------------------------------------

__global__ void zero_kernel(float* __restrict__ p, int n) {
    int i = blockIdx.x * blockDim.x + threadIdx.x;
    if (i < n) p[i] = 0.0f;
}

// ---------------- WMMA GEMM: C[M,128] = A[M,128] * W[128,128] ---------------
// One wave per 16-row strip; wave covers all 8 column tiles (A reused 8x per
// k-step). W staged in LDS (64 KB). f32 WMMA 16x16x4, 32 k-steps.
//
// Per CDNA5 ISA layouts (CDNA5 ISA):
//   A 16x4:  VGPR0 = K=k (lanes 0-15) / K=k+2 (lanes 16-31); VGPR1 = k+1/k+3
//   B 4x16:  VGPR0 = K=k (lanes 0-15, N=lane) / K=k+2 (lanes 16-31, N=lane-16)
//   C 16x16: VGPR v = M=v (lanes 0-15) / M=v+8 (lanes 16-31), N=lane%16

__global__ __launch_bounds__(256)
void gemm128_wmma_kernel(const float* __restrict__ A, const float* __restrict__ W,
                         float* __restrict__ C, int M) {
    __shared__ float lW[F_DIM * F_DIM];   // 64 KB of the 320 KB WGP LDS
    for (int i = threadIdx.x; i < F_DIM * F_DIM; i += 256) lW[i] = W[i];
    __syncthreads();

    const int wave = threadIdx.x >> 5;
    const int lane = threadIdx.x & 31;
    const int strip = blockIdx.x * 8 + wave;       // 16 rows per strip
    const int row0  = strip * 16;
    if (row0 >= M) return;                          // wave-uniform guard

    const int half = lane >> 4;                     // 0: lanes 0-15, 1: 16-31
    const int l16  = lane & 15;

    const v8f zero = {0.f, 0.f, 0.f, 0.f, 0.f, 0.f, 0.f, 0.f};
    v8f acc[8];
#pragma unroll
    for (int t = 0; t < 8; ++t) acc[t] = zero;

    const float* __restrict__ arow = A + (size_t)(row0 + l16) * F_DIM;

    for (int k0 = 0; k0 < F_DIM; k0 += 4) {
        const int ka = k0 + half * 2;               // this lane's K pair
        v2f a;
        a.x = arow[ka];
        a.y = arow[ka + 1];
#pragma unroll
        for (int t = 0; t < 8; ++t) {
            const int col = t * 16 + l16;
            v2f b;
            b.x = lW[ka * F_DIM + col];
            b.y = lW[(ka + 1) * F_DIM + col];
            acc[t] = __builtin_amdgcn_wmma_f32_16x16x4_f32(
                /*neg_a=*/false, a, /*neg_b=*/false, b,
                /*c_mod=*/(short)0, acc[t], /*reuse_a=*/false, /*reuse_b=*/false);
        }
    }

    // Write back per C/D layout.
#pragma unroll
    for (int t = 0; t < 8; ++t) {
        const int col = t * 16 + l16;
#pragma unroll
        for (int v = 0; v < 8; ++v) {
            C[(size_t)(row0 + v + 8 * half) * F_DIM + col] = acc[t][v];
        }
    }
}

// ---------------- edge scatter: out[dst] += H[src] * dis[src]*dis[dst] ------
// One wave per edge; lane handles 4 consecutive features (float4 gather,
// 4x global_atomic_add_f32 scatter). 512 B coalesced per edge gather.

__global__ void scatter_kernel(const float* __restrict__ H,
                               const int* __restrict__ src,
                               const int* __restrict__ dst,
                               const float* __restrict__ dis,
                               float* __restrict__ out, int e) {
    int tid  = blockIdx.x * blockDim.x + threadIdx.x;
    int edge = tid >> 5;
    int lane = tid & 31;
    if (edge >= e) return;
    int s = src[edge];
    int d = dst[edge];
    float norm = dis[s] * dis[d];
    const float4* __restrict__ hv = (const float4*)(H + (size_t)s * F_DIM);
    float4 m = hv[lane];
    float* o = out + (size_t)d * F_DIM + lane * 4;
    atomicAdd(o + 0, m.x * norm);
    atomicAdd(o + 1, m.y * norm);
    atomicAdd(o + 2, m.z * norm);
    atomicAdd(o + 3, m.w * norm);
}

// ---------------- fused self-loop + bias + ReLU (in place on S) -------------
// S[i,h] = relu(S[i,h] + H[i,h]*dis[i]^2 + b[h])

__global__ void selfloop_bias_relu_kernel(float* __restrict__ S,
                                          const float* __restrict__ H,
                                          const float* __restrict__ dis,
                                          const float* __restrict__ b, int n) {
    int tid = blockIdx.x * blockDim.x + threadIdx.x;   // n * 128 threads
    if (tid >= n * F_DIM) return;
    int i = tid >> 7;
    int h = tid & (F_DIM - 1);
    float di = dis[i];
    float v = S[tid] + H[tid] * di * di + b[h];
    S[tid] = fmaxf(v, 0.0f);
}

// ---------------- global add pool -------------------------------------------

__global__ void pool_kernel(const float* __restrict__ H,
                            const int* __restrict__ batch,
                            float* __restrict__ pooled, int n) {
    int tid  = blockIdx.x * blockDim.x + threadIdx.x;
    int node = tid >> 5;
    int lane = tid & 31;
    if (node >= n) return;
    int g = batch[node];
    const float4* __restrict__ hv = (const float4*)(H + (size_t)node * F_DIM);
    float4 m = hv[lane];
    float* o = pooled + (size_t)g * F_DIM + lane * 4;
    atomicAdd(o + 0, m.x);
    atomicAdd(o + 1, m.y);
    atomicAdd(o + 2, m.z);
    atomicAdd(o + 3, m.w);
}

// ---------------- head: logits + log_softmax --------------------------------
// One thread per graph: 16 classes x 128 MACs, then row log_softmax.

__global__ void head_kernel(const float* __restrict__ pooled,
                            const float* __restrict__ Wlin,
                            const float* __restrict__ blin,
                            float* __restrict__ out, int g_total) {
    int g = blockIdx.x * blockDim.x + threadIdx.x;
    if (g >= g_total) return;
    float logits[N_CLASSES];
#pragma unroll
    for (int c = 0; c < N_CLASSES; ++c) logits[c] = blin[c];
    const float* __restrict__ pr = pooled + (size_t)g * F_DIM;
    for (int k = 0; k < F_DIM; ++k) {
        float p = pr[k];
        const float* __restrict__ wr = Wlin + (size_t)k * N_CLASSES;
#pragma unroll
        for (int c = 0; c < N_CLASSES; ++c) logits[c] = fmaf(p, wr[c], logits[c]);
    }
    float mx = logits[0];
#pragma unroll
    for (int c = 1; c < N_CLASSES; ++c) mx = fmaxf(mx, logits[c]);
    float s = 0.0f;
#pragma unroll
    for (int c = 0; c < N_CLASSES; ++c) s += expf(logits[c] - mx);
    float lse = mx + logf(s);
    float* o = out + (size_t)g * N_CLASSES;
#pragma unroll
    for (int c = 0; c < N_CLASSES; ++c) o[c] = logits[c] - lse;
}

// ---------------------------------------------------------------------------

extern "C" void kernel_launch(void* const* d_in, const int* in_sizes, int n_in,
                              void* d_out, int out_size, void* d_ws, size_t ws_size,
                              hipStream_t stream) {
    const float* x     = (const float*)d_in[0];
    const int*   eidx  = (const int*)d_in[1];       // [2, E] flat
    const int*   batch = (const int*)d_in[2];
    const float* W1    = (const float*)d_in[3];
    const float* b1    = (const float*)d_in[4];
    const float* W2    = (const float*)d_in[5];
    const float* b2    = (const float*)d_in[6];
    const float* Wlin  = (const float*)d_in[7];
    const float* blin  = (const float*)d_in[8];
    float*       out   = (float*)d_out;

    const int* src = eidx;
    const int* dst = eidx + N_EDGES;

    // workspace layout (floats)
    float* ws = (float*)d_ws;
    float* deg    = ws;                                      //  50000
    float* dis    = ws + 50000;                              //  50000
    float* H      = ws + 100000;                             //  50000*128
    float* S      = H + (size_t)N_NODES * F_DIM;             //  50000*128
    float* pooled = S + (size_t)N_NODES * F_DIM;             //  500*128

    const int B = 256;
    const int nodeBlocks  = (N_NODES + B - 1) / B;
    const int edgeBlocks  = (N_EDGES + B - 1) / B;
    const int featBlocks  = ((size_t)N_NODES * F_DIM + B - 1) / B;
    const int scatBlocks  = ((size_t)N_EDGES * 32 + B - 1) / B;
    const int poolBlocks  = ((size_t)N_NODES * 32 + B - 1) / B;
    const int gemmBlocks  = (N_NODES / 16 + 7) / 8;          // 8 waves/block
    const int headBlocks  = (N_GRAPHS + B - 1) / B;

    // --- normalization ---
    init_deg_kernel<<<nodeBlocks, B, 0, stream>>>(deg, N_NODES);
    deg_edges_kernel<<<edgeBlocks, B, 0, stream>>>(dst, deg, N_EDGES);
    dis_kernel<<<nodeBlocks, B, 0, stream>>>(deg, dis, N_NODES);

    // --- conv 1 ---
    gemm128_wmma_kernel<<<gemmBlocks, B, 0, stream>>>(x, W1, H, N_NODES);
    zero_kernel<<<featBlocks, B, 0, stream>>>(S, N_NODES * F_DIM);
    scatter_kernel<<<scatBlocks, B, 0, stream>>>(H, src, dst, dis, S, N_EDGES);
    selfloop_bias_relu_kernel<<<featBlocks, B, 0, stream>>>(S, H, dis, b1, N_NODES);

    // --- conv 2 (S holds relu1; GEMM2 into H, then reuse S as accumulator) ---
    gemm128_wmma_kernel<<<gemmBlocks, B, 0, stream>>>(S, W2, H, N_NODES);
    zero_kernel<<<featBlocks, B, 0, stream>>>(S, N_NODES * F_DIM);
    scatter_kernel<<<scatBlocks, B, 0, stream>>>(H, src, dst, dis, S, N_EDGES);
    selfloop_bias_relu_kernel<<<featBlocks, B, 0, stream>>>(S, H, dis, b2, N_NODES);

    // --- pool + head ---
    zero_kernel<<<(N_GRAPHS * F_DIM + B - 1) / B, B, 0, stream>>>(pooled, N_GRAPHS * F_DIM);
    pool_kernel<<<poolBlocks, B, 0, stream>>>(S, batch, pooled, N_NODES);
    head_kernel<<<headBlocks, B, 0, stream>>>(pooled, Wlin, blin, out, N_GRAPHS);
}